// ContrastiveAttention_17549236371713
// MI455X (gfx1250) — compile-verified
//
#include <hip/hip_runtime.h>
#include <math.h>

typedef __attribute__((ext_vector_type(2))) float v2f;
typedef __attribute__((ext_vector_type(4))) float v4f;
typedef __attribute__((ext_vector_type(8))) float v8f;

#define D_MODEL 512
#define D_FC    2048
#define POOL_N  100
#define ROUNDS  3
#define BATCH   512
#define SEQ_N   196
#define SCALE   0.044194173824159216f  /* 1/sqrt(512) */

// ---------------------------------------------------------------------------
// Generic f32 WMMA GEMM:  C[M x N] = A[M x K] * B[N x K]^T (+bias, opt ReLU)
// One wave (32 lanes) computes one 16x16 C tile via V_WMMA_F32_16X16X4_F32.
// grid.x = M/16, grid.y = N/16, grid.z = batch; blockDim = 32 (EXEC all-ones).
// ---------------------------------------------------------------------------
__global__ __launch_bounds__(32)
void wmma_gemm_f32(const float* __restrict__ A, const float* __restrict__ B,
                   const float* __restrict__ bias, float* __restrict__ C,
                   int K, int ldA, int ldB, int ldC,
                   long aStride, long bStride, long cStride, int relu)
{
    const int lane = threadIdx.x;
    const int half = lane >> 4;   // which K-pair (A) / K-pair (B)
    const int l16  = lane & 15;   // M row for A frag, N col for B frag

    const float* Ab = A + (size_t)blockIdx.z * aStride
                        + (size_t)(blockIdx.x * 16 + l16) * ldA;
    const float* Bb = B + (size_t)blockIdx.z * bStride
                        + (size_t)(blockIdx.y * 16 + l16) * ldB;

    v8f acc = {};
    #pragma unroll 4
    for (int k0 = 0; k0 < K; k0 += 4) {
        const int ka = k0 + 2 * half;
        v2f a = *(const v2f*)(Ab + ka);   // A[m][ka], A[m][ka+1]
        v2f b = *(const v2f*)(Bb + ka);   // B^T[ka][n], B^T[ka+1][n]
        acc = __builtin_amdgcn_wmma_f32_16x16x4_f32(
                  false, a, false, b, (short)0, acc, false, false);
    }

    float* Cb = C + (size_t)blockIdx.z * cStride;
    const int n  = blockIdx.y * 16 + l16;
    const float bv = bias ? bias[n] : 0.0f;
    #pragma unroll
    for (int i = 0; i < 8; ++i) {
        const int m = blockIdx.x * 16 + 8 * half + i;
        float v = acc[i] + bv;
        if (relu) v = fmaxf(v, 0.0f);
        Cb[(size_t)m * ldC + n] = v;
    }
}

// ---------------------------------------------------------------------------
// Per-b: scores = q_all[r,b,:] . pool[p,:] * scale ; softmax over p ;
// v_agg[b,:] = sum_r softmax . pool / ROUNDS.   128 threads per block.
// ---------------------------------------------------------------------------
__global__ __launch_bounds__(128)
void attn_pool_kernel(const float* __restrict__ q_all,
                      const float* __restrict__ pool,
                      float* __restrict__ v_agg)
{
    const int b = blockIdx.x;
    const int tid = threadIdx.x;
    __shared__ float qsh[D_MODEL];
    __shared__ float wsh[128];
    __shared__ float red[128];

    float acc0 = 0.f, acc1 = 0.f, acc2 = 0.f, acc3 = 0.f;

    for (int r = 0; r < ROUNDS; ++r) {
        const float* qr = q_all + ((size_t)r * BATCH + b) * D_MODEL;
        for (int i = tid; i < D_MODEL; i += 128) qsh[i] = qr[i];
        __syncthreads();

        float s = -1e30f;
        if (tid < POOL_N) {
            const float* pr = pool + (size_t)tid * D_MODEL;
            float d = 0.f;
            #pragma unroll 8
            for (int i = 0; i < D_MODEL; ++i) d += qsh[i] * pr[i];
            s = d * SCALE;
        }
        red[tid] = s;
        __syncthreads();
        for (int st = 64; st > 0; st >>= 1) {
            if (tid < st) red[tid] = fmaxf(red[tid], red[tid + st]);
            __syncthreads();
        }
        const float mx = red[0];
        __syncthreads();

        const float e = (tid < POOL_N) ? __expf(s - mx) : 0.f;
        red[tid] = e;
        __syncthreads();
        for (int st = 64; st > 0; st >>= 1) {
            if (tid < st) red[tid] += red[tid + st];
            __syncthreads();
        }
        const float inv = 1.0f / red[0];
        __syncthreads();
        wsh[tid] = e * inv;
        __syncthreads();

        // accumulate weights . pool  (coalesced over d)
        float a0 = 0.f, a1 = 0.f, a2 = 0.f, a3 = 0.f;
        for (int p = 0; p < POOL_N; ++p) {
            const float w = wsh[p];
            const float* pr = pool + (size_t)p * D_MODEL;
            a0 += w * pr[tid];
            a1 += w * pr[tid + 128];
            a2 += w * pr[tid + 256];
            a3 += w * pr[tid + 384];
        }
        acc0 += a0; acc1 += a1; acc2 += a2; acc3 += a3;
        __syncthreads();
    }

    float* vb = v_agg + (size_t)b * D_MODEL;
    const float k = 1.0f / (float)ROUNDS;
    vb[tid]       = acc0 * k;
    vb[tid + 128] = acc1 * k;
    vb[tid + 256] = acc2 * k;
    vb[tid + 384] = acc3 * k;
}

// ---------------------------------------------------------------------------
// Per-b: sim = (q.k)*scale ; tmp = v_hat - sigmoid(sim)*v
// ---------------------------------------------------------------------------
__global__ __launch_bounds__(256)
void sim_tmp_kernel(const float* __restrict__ q, const float* __restrict__ k,
                    const float* __restrict__ v, const float* __restrict__ v_hat,
                    float* __restrict__ tmp)
{
    const int b = blockIdx.x;
    const int tid = threadIdx.x;
    __shared__ float red[256];
    const float* qb = q + (size_t)b * D_MODEL;
    const float* kb = k + (size_t)b * D_MODEL;

    red[tid] = qb[tid] * kb[tid] + qb[tid + 256] * kb[tid + 256];
    __syncthreads();
    for (int st = 128; st > 0; st >>= 1) {
        if (tid < st) red[tid] += red[tid + st];
        __syncthreads();
    }
    const float s = 1.0f / (1.0f + __expf(-red[0] * SCALE));

    const float* vh = v_hat + (size_t)b * D_MODEL;
    const float* vb = v + (size_t)b * D_MODEL;
    float* tb = tmp + (size_t)b * D_MODEL;
    tb[tid]       = vh[tid]       - s * vb[tid];
    tb[tid + 256] = vh[tid + 256] - s * vb[tid + 256];
}

// ---------------------------------------------------------------------------
// Per-b: gpart[b] = v_diff[b,:] . gw_diff + gb
// ---------------------------------------------------------------------------
__global__ __launch_bounds__(256)
void gpart_kernel(const float* __restrict__ v_diff, const float* __restrict__ gw,
                  const float* __restrict__ gb, float* __restrict__ gpart)
{
    const int b = blockIdx.x;
    const int tid = threadIdx.x;
    __shared__ float red[256];
    const float* vd = v_diff + (size_t)b * D_MODEL;
    const float* gd = gw + D_MODEL;   // gw[:, 512:1024]

    red[tid] = vd[tid] * gd[tid] + vd[tid + 256] * gd[tid + 256];
    __syncthreads();
    for (int st = 128; st > 0; st >>= 1) {
        if (tid < st) red[tid] += red[tid + st];
        __syncthreads();
    }
    if (tid == 0) gpart[b] = red[0] + gb[0];
}

// ---------------------------------------------------------------------------
// Streaming epilogue (bandwidth-roofline kernel):
// out[b,n,:] = att + rs * sigmoid(att[b,n,:].gw_att + gpart[b]) * v_diff[b,:]
// grid = (SEQ_N, BATCH), 128 threads, one native float4 (v4f) per thread.
// att is read once (NT load) and out written once (NT store) so the 410 MB
// stream does not evict the L2-resident small tensors (v_diff/gw/gpart).
// Gate reduction: wave32 __shfl_xor butterfly + one 4-float LDS exchange.
// ---------------------------------------------------------------------------
__global__ __launch_bounds__(128)
void final_kernel(const float* __restrict__ att, const float* __restrict__ v_diff,
                  const float* __restrict__ gpart, const float* __restrict__ gw,
                  const float* __restrict__ rs, float* __restrict__ out)
{
    const int n = blockIdx.x;
    const int b = blockIdx.y;
    const int tid = threadIdx.x;
    __shared__ float wsum[4];

    const size_t row = ((size_t)b * SEQ_N + n) * D_MODEL;
    const v4f a4 = __builtin_nontemporal_load((const v4f*)(att + row) + tid);
    const v4f g4 = ((const v4f*)gw)[tid];   // gw_att = gw[:, :512]

    float p = a4.x * g4.x + a4.y * g4.y + a4.z * g4.z + a4.w * g4.w;
    // wave32 butterfly reduction
    #pragma unroll
    for (int off = 16; off > 0; off >>= 1)
        p += __shfl_xor(p, off, 32);
    if ((tid & 31) == 0) wsum[tid >> 5] = p;
    __syncthreads();
    const float logit = wsum[0] + wsum[1] + wsum[2] + wsum[3] + gpart[b];

    const float g = 1.0f / (1.0f + __expf(-logit));
    const float c = rs[0] * g;

    const v4f vd = ((const v4f*)(v_diff + (size_t)b * D_MODEL))[tid];
    v4f o;
    o.x = a4.x + c * vd.x;
    o.y = a4.y + c * vd.y;
    o.z = a4.z + c * vd.z;
    o.w = a4.w + c * vd.w;
    __builtin_nontemporal_store(o, (v4f*)(out + row) + tid);
}

// ---------------------------------------------------------------------------
extern "C" void kernel_launch(void* const* d_in, const int* in_sizes, int n_in,
                              void* d_out, int out_size, void* d_ws, size_t ws_size,
                              hipStream_t stream)
{
    const float* att_feats = (const float*)d_in[0];   // (512,196,512)
    const float* fc_feats  = (const float*)d_in[1];   // (512,2048)
    const float* pool      = (const float*)d_in[2];   // (100,512)
    const float* fc_proj_w = (const float*)d_in[3];   // (512,2048)
    const float* agg_w     = (const float*)d_in[4];   // (3,512,512)
    const float* qw        = (const float*)d_in[5];
    const float* qb        = (const float*)d_in[6];
    const float* kw        = (const float*)d_in[7];
    const float* kb        = (const float*)d_in[8];
    const float* vw        = (const float*)d_in[9];
    const float* vb        = (const float*)d_in[10];
    const float* cw        = (const float*)d_in[11];
    const float* cb        = (const float*)d_in[12];
    const float* gw        = (const float*)d_in[13];  // (1,1024)
    const float* gb        = (const float*)d_in[14];
    const float* rs        = (const float*)d_in[15];
    float* out = (float*)d_out;

    // workspace carve-up (floats)
    float* ws      = (float*)d_ws;
    const size_t T = (size_t)BATCH * D_MODEL;  // 262144
    float* v_hat  = ws;                 // 1 MB
    float* q_all  = v_hat  + T;         // 3 MB
    float* v_agg  = q_all  + 3 * T;
    float* qbuf   = v_agg  + T;
    float* kbuf   = qbuf   + T;
    float* vbuf   = kbuf   + T;
    float* tmp    = vbuf   + T;
    float* v_diff = tmp    + T;
    float* gpart  = v_diff + T;         // 512 floats

    const dim3 wblk(32, 1, 1);

    // 1. v_hat = fc_feats @ fc_proj_w^T   (512x512, K=2048)
    wmma_gemm_f32<<<dim3(BATCH / 16, D_MODEL / 16, 1), wblk, 0, stream>>>(
        fc_feats, fc_proj_w, nullptr, v_hat,
        D_FC, D_FC, D_FC, D_MODEL, 0, 0, 0, 0);

    // 2. q_all[r] = v_hat @ agg_w[r]^T   (batched over r)
    wmma_gemm_f32<<<dim3(BATCH / 16, D_MODEL / 16, ROUNDS), wblk, 0, stream>>>(
        v_hat, agg_w, nullptr, q_all,
        D_MODEL, D_MODEL, D_MODEL, D_MODEL,
        0, (long)D_MODEL * D_MODEL, (long)BATCH * D_MODEL, 0);

    // 3. softmax pooling -> v_agg
    attn_pool_kernel<<<BATCH, 128, 0, stream>>>(q_all, pool, v_agg);

    // 4. q / k / v projections
    wmma_gemm_f32<<<dim3(BATCH / 16, D_MODEL / 16, 1), wblk, 0, stream>>>(
        v_hat, qw, qb, qbuf, D_MODEL, D_MODEL, D_MODEL, D_MODEL, 0, 0, 0, 0);
    wmma_gemm_f32<<<dim3(BATCH / 16, D_MODEL / 16, 1), wblk, 0, stream>>>(
        v_agg, kw, kb, kbuf, D_MODEL, D_MODEL, D_MODEL, D_MODEL, 0, 0, 0, 0);
    wmma_gemm_f32<<<dim3(BATCH / 16, D_MODEL / 16, 1), wblk, 0, stream>>>(
        v_agg, vw, vb, vbuf, D_MODEL, D_MODEL, D_MODEL, D_MODEL, 0, 0, 0, 0);

    // 5. tmp = v_hat - sigmoid(q.k * scale) * v
    sim_tmp_kernel<<<BATCH, 256, 0, stream>>>(qbuf, kbuf, vbuf, v_hat, tmp);

    // 6. v_diff = relu(tmp @ cw^T + cb)
    wmma_gemm_f32<<<dim3(BATCH / 16, D_MODEL / 16, 1), wblk, 0, stream>>>(
        tmp, cw, cb, v_diff, D_MODEL, D_MODEL, D_MODEL, D_MODEL, 0, 0, 0, 1);

    // 7. gpart[b] = v_diff . gw_diff + gb
    gpart_kernel<<<BATCH, 256, 0, stream>>>(v_diff, gw, gb, gpart);

    // 8. streaming gated-residual epilogue (bandwidth bound)
    final_kernel<<<dim3(SEQ_N, BATCH, 1), 128, 0, stream>>>(
        att_feats, v_diff, gpart, gw, rs, out);
}